// MLPPredictor_70153995813483
// MI455X (gfx1250) — compile-verified
//
#include <hip/hip_runtime.h>
#include <hip/hip_bf16.h>

typedef __attribute__((ext_vector_type(2))) float v2f;
typedef __attribute__((ext_vector_type(8))) float v8f;

#define E_TOTAL 625000
#define DFEAT   128
#define NTILES  ((E_TOTAL + 15) / 16)     // 39063 tiles of 16 edges
#define WAVES_PER_BLOCK 8
#define THREADS (WAVES_PER_BLOCK * 32)

// One wave32 per 16-edge tile.
// score[e][n] = sum_f h[e][f] * W[n][f] + b[n],  h = [nfeats[src] | nfeats[dst] | efeats]
// Mapped to V_WMMA_F32_16X16X4_F32: A = 16 edges x 4 feats, B = 4 feats x 16 (N padded, 2 valid),
// accumulated over 96 K-chunks in an 8-VGPR f32 accumulator.
__global__ __launch_bounds__(THREADS) void mlp_edge_score_wmma(
    const float* __restrict__ nfeats,   // [N_NODES, 128]
    const float* __restrict__ efeats,   // [E, 128]
    const int*   __restrict__ src,      // [E]
    const int*   __restrict__ dst,      // [E]
    const float* __restrict__ W,        // [2, 384]
    const float* __restrict__ bias,     // [2]
    float*       __restrict__ out)      // [E, 2]
{
    // Pre-expanded B fragments: for chunk k, lane L holds B-pair
    // (W[n][4k+2H], W[n][4k+2H+1]) with n = L%16 (zero for n>=2), H = L/16.
    __shared__ float ldsB[96 * 32 * 2];   // 24 KB

    const int tid = threadIdx.x;
    for (int idx = tid; idx < 96 * 32; idx += THREADS) {
        const int k = idx >> 5;
        const int L = idx & 31;
        const int n = L & 15;
        const int H = L >> 4;
        const int f = 4 * k + 2 * H;
        float b0 = 0.0f, b1 = 0.0f;
        if (n < 2) {
            b0 = W[n * 384 + f];
            b1 = W[n * 384 + f + 1];
        }
        ldsB[idx * 2 + 0] = b0;
        ldsB[idx * 2 + 1] = b1;
    }
    __syncthreads();

    const int wave = tid >> 5;
    const int lane = tid & 31;
    const int tile = blockIdx.x * WAVES_PER_BLOCK + wave;
    if (tile >= NTILES) return;          // wave-uniform: EXEC stays all-ones for WMMA

    const int m = lane & 15;             // A-matrix row (edge within tile)
    const int H = lane >> 4;             // lane half -> K sub-offset
    const int e  = tile * 16 + m;
    const int eL = (e < E_TOTAL) ? e : (E_TOTAL - 1);   // clamp loads on tail tile

    // Per-lane base pointers for the three 128-float segments; +2H folds the
    // lane-half K offset so the unrolled loop uses pure immediate offsets.
    const float* pS = nfeats + (size_t)src[eL] * DFEAT + 2 * H;
    const float* pD = nfeats + (size_t)dst[eL] * DFEAT + 2 * H;
    const float* pE = efeats + (size_t)eL      * DFEAT + 2 * H;
    const v2f*   pB = (const v2f*)ldsB + lane;          // chunk stride = 32 v2f

    v8f c = {};
#pragma unroll
    for (int k = 0; k < 32; ++k) {
        v2f a = *(const v2f*)(pS + 4 * k);
        v2f b = pB[k * 32];
        c = __builtin_amdgcn_wmma_f32_16x16x4_f32(false, a, false, b,
                                                  (short)0, c, false, false);
    }
#pragma unroll
    for (int k = 0; k < 32; ++k) {
        v2f a = *(const v2f*)(pD + 4 * k);
        v2f b = pB[(32 + k) * 32];
        c = __builtin_amdgcn_wmma_f32_16x16x4_f32(false, a, false, b,
                                                  (short)0, c, false, false);
    }
#pragma unroll
    for (int k = 0; k < 32; ++k) {
        v2f a = *(const v2f*)(pE + 4 * k);
        v2f b = pB[(64 + k) * 32];
        c = __builtin_amdgcn_wmma_f32_16x16x4_f32(false, a, false, b,
                                                  (short)0, c, false, false);
    }

    // D layout: VGPR v, lane-half H, lane n (= m) -> score[tile*16 + v + 8H][n].
    // Only N=0,1 columns are real outputs.
    if (m < 2) {
        const float bv = bias[m];
#pragma unroll
        for (int v = 0; v < 8; ++v) {
            const int eo = tile * 16 + v + 8 * H;
            if (eo < E_TOTAL)
                out[(size_t)eo * 2 + m] = c[v] + bv;
        }
    }
}

extern "C" void kernel_launch(void* const* d_in, const int* in_sizes, int n_in,
                              void* d_out, int out_size, void* d_ws, size_t ws_size,
                              hipStream_t stream) {
    const float* nfeats = (const float*)d_in[0];
    const float* efeats = (const float*)d_in[1];
    const int*   src    = (const int*)  d_in[2];
    const int*   dst    = (const int*)  d_in[3];
    const float* W      = (const float*)d_in[4];
    const float* bias   = (const float*)d_in[5];
    float*       out    = (float*)d_out;

    const int blocks = (NTILES + WAVES_PER_BLOCK - 1) / WAVES_PER_BLOCK;
    mlp_edge_score_wmma<<<blocks, THREADS, 0, stream>>>(
        nfeats, efeats, src, dst, W, bias, out);
}